// Project_3444563772240
// MI455X (gfx1250) — compile-verified
//
#include <hip/hip_runtime.h>
#include <hip/hip_bf16.h>
#include <math.h>
#include <stdint.h>

typedef __attribute__((ext_vector_type(2))) float    v2f;
typedef __attribute__((ext_vector_type(8))) float    v8f;
typedef __attribute__((ext_vector_type(4))) unsigned u32x4;
typedef __attribute__((ext_vector_type(8))) unsigned u32x8;

#define NX      1000
#define NXX     (NX * NX)
#define NGS     25
#define NG      40            // spatial tiles per dim
#define KDIM    64            // nb * nt
#define PDIM    625           // pixels per tile
#define WVAL    75            // valid columns per chunk (3 tile rows)
#define WS      82            // Vc row stride (floats): even (b64 align) + bank-conflict free
#define XSTR    80            // Xs row stride (floats)
#define NCHUNK  9             // ceil(625/75)
#define KSTEP   19            // k-steps of 4 covering cols 0..75 (col 75 is zero pad)
#define DAMP    0.01f

// LDS layout (float offsets) — 15808 floats = 63232 B (< 64 KB static)
#define OFF_VC0 0                         // 64 x 82 = 5248
#define OFF_VC1 (OFF_VC0 + KDIM * WS)     // 5248
#define OFF_XS0 (OFF_VC1 + KDIM * WS)     // 4 x 80 = 320
#define OFF_XS1 (OFF_XS0 + 4 * XSTR)      // 320
#define OFF_G   (OFF_XS1 + 4 * XSTR)      // 64 x 64 = 4096
#define OFF_A   (OFF_G  + KDIM * KDIM)    // 4 x 64
#define OFF_B   (OFF_A  + 4 * KDIM)       // 4 x 64
#define OFF_DI  (OFF_B  + 4 * KDIM)       // 64
#define SMEM_F  (OFF_DI + KDIM)
// Permanently-zero 8-byte slot: Xs row pads (cols 75..79) are never written by the TDM.
#define OFF_Z2  (OFF_XS0 + 76)

// ---- Tensor Data Mover: 2-D tile (25 x 3 rows) iterated `iters` times ----
// D# per CDNA5 ISA §8: group0 = {flags(count=1), lds_addr, global_addr, type=2}
// group1 = {data_size=4B | iterate_en, tensor_dim0=25, tensor_dim1=rowsRem,
//           tile_dim0=25, tile_dim1=3, dim0_stride=1000}
// group2 = {-, lds_addr_increment, global_addr_increment=1e6, iterate_count}
// OOB rows (rowsRem < 3 on the last chunk) are written as zeros by the TDM.
__device__ __forceinline__ void tdm_tile_load(const float* gsrc, const float* ldsDst,
                                              int rowsRem, unsigned ldsInc, int iters)
{
    uint64_t ga = (uint64_t)(uintptr_t)gsrc;
    unsigned lo = (unsigned)(uintptr_t)ldsDst;   // flat addr low 32 bits == LDS byte offset
    u32x4 g0 = { 1u, lo, (unsigned)ga,
                 (unsigned)((ga >> 32) & 0x01FFFFFFu) | (2u << 30) };
    u32x8 g1 = { (2u << 16) | (1u << 19),        // data_size=2(4B), iterate_enable=1
                 25u << 16,                      // tensor_dim0 = 25
                 (unsigned)rowsRem << 16,        // tensor_dim1 = remaining tile rows
                 25u << 16,                      // tile_dim0 = 25
                 3u,                             // tile_dim1 = 3, tile_dim2 = 0
                 1000u,                          // tensor_dim0_stride
                 0u, 0u };
    u32x4 g2 = { 0u, ldsInc, 1000000u, (unsigned)(iters - 1) << 16 };
    u32x4 g3 = { 0u, 0u, 0u, 0u };
    asm volatile("tensor_load_to_lds %0, %1, %2, %3"
                 :: "s"(g0), "s"(g1), "s"(g2), "s"(g3) : "memory");
}

__device__ __forceinline__ v8f wmma_f32(v2f a, v2f b, v8f c) {
    return __builtin_amdgcn_wmma_f32_16x16x4_f32(false, a, false, b,
                                                 (short)0, c, false, false);
}

__global__ __launch_bounds__(256) void project_tile_kernel(
    const float* __restrict__ Xg,   // local_x   (4, 1000, 1000)
    const float* __restrict__ Vg,   // ms_history(64, 1000, 1000)
    float* __restrict__ Wout)       // out       (4, 1000, 1000)
{
    __shared__ float sm[SMEM_F];
    float* G  = sm + OFF_G;
    float* A4 = sm + OFF_A;
    float* B4 = sm + OFF_B;
    float* Di = sm + OFF_DI;

    const int tid  = threadIdx.x;
    const int lane = tid & 31;
    const int wav  = tid >> 5;
    const int tile = blockIdx.x;
    const int gx   = tile / NG;
    const int gy   = tile - gx * NG;
    const int bx   = gx * NGS * NX + gy * NGS;   // tile origin inside a 1000x1000 image

    const int mrow = lane & 15;
    const int koff = (lane < 16) ? 0 : 2;        // k-pair selector (A/B operand layout)
    const int rbase = (lane >> 4) << 3;          // C/D: lanes 16-31 hold M = r + 8

    // zero V/X staging buffers once (covers stride pads incl. the OFF_Z2 zero slot)
    for (int idx = tid; idx < OFF_G; idx += 256) sm[idx] = 0.0f;
    __syncthreads();

    // ---------- Phase 1: G = V V^T + damp*I  and  A = X V^T ----------
    const int ti1 = wav >> 2, tj = wav & 3, ti2 = ti1 + 2;
    v8f accG0 = {}, accG1 = {}, accA = {};

    if (wav == 0) {   // kick off chunk 0 DMA (X then V, in-order per wave)
        tdm_tile_load(Xg + bx, sm + OFF_XS0, 25, XSTR, 4);
        tdm_tile_load(Vg + bx, sm + OFF_VC0, 25, WS, KDIM);
    }

    for (int ch = 0; ch < NCHUNK; ++ch) {
        const int buf = ch & 1;
        if (wav == 0) {
            if (ch + 1 < NCHUNK) {   // prefetch next chunk into the other buffer
                const int r0 = (ch + 1) * 3;
                tdm_tile_load(Xg + bx + r0 * NX,
                              sm + (buf ? OFF_XS0 : OFF_XS1), 25 - r0, XSTR, 4);
                tdm_tile_load(Vg + bx + r0 * NX,
                              sm + (buf ? OFF_VC0 : OFF_VC1), 25 - r0, WS, KDIM);
                __builtin_amdgcn_s_wait_tensorcnt(2);   // current chunk's pair done
            } else {
                __builtin_amdgcn_s_wait_tensorcnt(0);
            }
        }
        __syncthreads();

        const float* VcB = sm + (buf ? OFF_VC1 : OFF_VC0);
        const int    xsB = buf ? OFF_XS1 : OFF_XS0;
        // branchless zero-padding: lanes with mrow>=4 read the permanent zero slot
        const int    xrow = (mrow < 4) ? (xsB + mrow * XSTR) : OFF_Z2;
        const int    xadd = (mrow < 4) ? 1 : 0;          // advance only real rows

        // software-pipelined fragments: load step ks+1, then issue step ks WMMAs
        v2f a1 = *(const v2f*)&VcB[(ti1 * 16 + mrow) * WS + koff];
        v2f a2 = *(const v2f*)&VcB[(ti2 * 16 + mrow) * WS + koff];
        v2f b  = *(const v2f*)&VcB[(tj  * 16 + mrow) * WS + koff];
        v2f x  = *(const v2f*)&sm[xrow + xadd * koff];

#pragma unroll 4
        for (int ks = 0; ks < KSTEP; ++ks) {
            const int np = (ks + 1 < KSTEP ? (ks + 1) * 4 : 0) + koff;
            v2f na1 = *(const v2f*)&VcB[(ti1 * 16 + mrow) * WS + np];
            v2f na2 = *(const v2f*)&VcB[(ti2 * 16 + mrow) * WS + np];
            v2f nb  = *(const v2f*)&VcB[(tj  * 16 + mrow) * WS + np];
            v2f nx  = *(const v2f*)&sm[xrow + xadd * np];

            accG0 = wmma_f32(a1, b, accG0);
            accG1 = wmma_f32(a2, b, accG1);
            accA  = wmma_f32(x,  b, accA);   // waves 4-7 compute a discarded duplicate;
                                             // keeps EXEC all-ones (ISA requirement)
            a1 = na1; a2 = na2; b = nb; x = nx;
        }
        __syncthreads();
    }

    // spill accumulators: C/D layout VGPR r -> (M = r | r+8, N = lane&15)
    {
        const int col = tj * 16 + mrow;
        for (int r = 0; r < 8; ++r) {
            int row1 = ti1 * 16 + rbase + r;
            int row2 = ti2 * 16 + rbase + r;
            G[row1 * KDIM + col] = accG0[r] + (row1 == col ? DAMP : 0.0f);
            G[row2 * KDIM + col] = accG1[r] + (row2 == col ? DAMP : 0.0f);
        }
        if (wav < 4 && lane < 16) {
            for (int r = 0; r < 4; ++r)
                A4[r * KDIM + tj * 16 + mrow] = accA[r];
        }
    }

    // ---------- Phase 2: Cholesky G = L L^T (in place, symmetric full update) ----------
    const int rq = tid >> 2;      // row slot     (0..63)
    const int rr = tid & 3;       // column phase (0..3)
    for (int j = 0; j < KDIM; ++j) {
        __syncthreads();
        if (tid == 0) {
            float d = sqrtf(G[j * KDIM + j]);
            G[j * KDIM + j] = d;
            Di[j] = 1.0f / d;
        }
        __syncthreads();
        for (int i = j + 1 + tid; i < KDIM; i += 256)
            G[i * KDIM + j] *= Di[j];
        __syncthreads();
        const int i = j + 1 + rq;               // each thread owns one trailing row
        if (i < KDIM) {
            const float lij = G[i * KDIM + j];
            for (int c = j + 1 + rr; c < KDIM; c += 4)
                G[i * KDIM + c] -= lij * G[c * KDIM + j];
        }
    }
    __syncthreads();

    // Z = A G^{-1} via L w = a^T then L^T z = w (wave r owns RHS r)
    for (int idx = tid; idx < 4 * KDIM; idx += 256) B4[idx] = A4[idx];
    for (int j = 0; j < KDIM; ++j) {
        __syncthreads();
        if (wav < 4) {
            float xj = B4[wav * KDIM + j] * Di[j];
            for (int i = j + 1 + lane; i < KDIM; i += 32)
                B4[wav * KDIM + i] -= G[i * KDIM + j] * xj;
            if (lane == 0) B4[wav * KDIM + j] = xj;
        }
    }
    for (int j = KDIM - 1; j >= 0; --j) {
        __syncthreads();
        if (wav < 4) {
            float zj = B4[wav * KDIM + j] * Di[j];
            for (int i = lane; i < j; i += 32)
                B4[wav * KDIM + i] -= G[j * KDIM + i] * zj;
            if (lane == 0) B4[wav * KDIM + j] = zj;
        }
    }
    __syncthreads();

    // ---------- Phase 3: Y = Z V, TDM-streamed chunks, double-buffered ----------
    if (wav == 0) tdm_tile_load(Vg + bx, sm + OFF_VC0, 25, WS, KDIM);

    // branchless Z-fragment addressing (rows >= 4 read the permanent zero slot)
    const int zrow = (mrow < 4) ? (OFF_B + mrow * KDIM) : OFF_Z2;
    const int zadd = (mrow < 4) ? 1 : 0;

    for (int ch = 0; ch < NCHUNK; ++ch) {
        const int buf = ch & 1;
        if (wav == 0) {
            if (ch + 1 < NCHUNK) {
                const int r0 = (ch + 1) * 3;
                tdm_tile_load(Vg + bx + r0 * NX,
                              sm + (buf ? OFF_VC0 : OFF_VC1), 25 - r0, WS, KDIM);
                __builtin_amdgcn_s_wait_tensorcnt(1);
            } else {
                __builtin_amdgcn_s_wait_tensorcnt(0);
            }
        }
        __syncthreads();

        const float* VcB = sm + (buf ? OFF_VC1 : OFF_VC0);
        if (wav < 5) {                      // 5 x 16-column subtiles cover 75 cols
            const int cb = wav * 16;
            v8f accY = {};
            v2f z = *(const v2f*)&sm[zrow + zadd * koff];
            v2f vv;
            vv.x = VcB[(koff    ) * WS + cb + mrow];
            vv.y = VcB[(koff + 1) * WS + cb + mrow];

#pragma unroll 4
            for (int ks = 0; ks < KDIM / 4; ++ks) {
                const int nk = (ks + 1 < KDIM / 4 ? (ks + 1) * 4 : 0) + koff;
                v2f nz = *(const v2f*)&sm[zrow + zadd * nk];
                v2f nv;
                nv.x = VcB[(nk    ) * WS + cb + mrow];
                nv.y = VcB[(nk + 1) * WS + cb + mrow];
                accY = wmma_f32(z, vv, accY);
                z = nz; vv = nv;
            }
            // real outputs are rows M=0..3: VGPRs 0..3, lanes 0..15
            if (lane < 16) {
                const int cl = cb + lane;            // chunk-local column
                const int gp = ch * WVAL + cl;       // global tile pixel
                if (cl < WVAL && gp < PDIM) {
                    const int off = bx + (gp / NGS) * NX + (gp % NGS);
                    for (int r = 0; r < 4; ++r)
                        Wout[r * NXX + off] = accY[r];
                }
            }
        }
        __syncthreads();
    }
}

extern "C" void kernel_launch(void* const* d_in, const int* in_sizes, int n_in,
                              void* d_out, int out_size, void* d_ws, size_t ws_size,
                              hipStream_t stream) {
    (void)in_sizes; (void)n_in; (void)d_ws; (void)ws_size; (void)out_size;
    const float* local_x    = (const float*)d_in[0];   // (1,4,1000,1000)
    const float* ms_history = (const float*)d_in[1];   // (1,16,4,1000,1000) == (64,1000,1000)
    float* out = (float*)d_out;                        // (1,4,1000,1000)
    project_tile_kernel<<<NG * NG, 256, 0, stream>>>(local_x, ms_history, out);
}